// Node2VecHypergraphConv_40638980555154
// MI455X (gfx1250) — compile-verified
//
#include <hip/hip_runtime.h>

// ---------------------------------------------------------------------------
// Node2Vec + HypergraphConv + Gram pipeline for MI455X (gfx1250, wave32).
//
// Pipeline (all f32, matching reference semantics exactly):
//   1. x = emb @ conv_w^T                      (WMMA f32 16x16x4, b64 loads)
//   2. D,B degree histograms                   (global atomics)
//   3. e += x[node_idx] scatter, e *= 1/B      (L2-resident f32 atomics)
//   4. n += e[edge_idx] scatter, y = leaky(n/D + cb)
//   5. g = y^T y  LDS-staged split-K strip GEMM (WMMA + ds loads + atomic red.)
//   6. out = leaky(g @ lin_w^T + lin_b)        (WMMA f32 16x16x4)
// ---------------------------------------------------------------------------

typedef __attribute__((ext_vector_type(2))) float v2f;
typedef __attribute__((ext_vector_type(8))) float v8f;

#define CH 256
#define NNODES 50000
#define NEDGES 10000
#define NNZ 800000
#define NEG_SLOPE 0.01f

// Gram split-K geometry: 50000 = 25 chunks * 2000 rows; LDS tile = 40 rows.
#define GRAM_CHUNKS 25
#define GRAM_KCHUNK 2000
#define GRAM_KT 40              // 40*256*4B = 40 KB LDS per block

__device__ __forceinline__ float leaky(float v) {
    return v >= 0.0f ? v : NEG_SLOPE * v;
}

// ---- zero workspace region -------------------------------------------------
__global__ void zero_f(float* __restrict__ p, long n) {
    long i = (long)blockIdx.x * blockDim.x + threadIdx.x;
    long stride = (long)gridDim.x * blockDim.x;
    for (; i < n; i += stride) p[i] = 0.0f;
}

// ---- degree histograms -----------------------------------------------------
__global__ __launch_bounds__(256) void degrees_k(const int* __restrict__ ni,
                                                 const int* __restrict__ ei,
                                                 float* __restrict__ D,
                                                 float* __restrict__ B) {
    int i = blockIdx.x * blockDim.x + threadIdx.x;
    if (i < NNZ) {
        atomicAdd(&D[ni[i]], 1.0f);
        atomicAdd(&B[ei[i]], 1.0f);
    }
}

// ---- x = emb @ conv_w^T  (one 16x16 tile per wave, K=256 in steps of 4) ----
__global__ __launch_bounds__(256) void gemm_xw_k(const float* __restrict__ emb,
                                                 const float* __restrict__ w,
                                                 float* __restrict__ x) {
    const int wave = blockIdx.x * (blockDim.x >> 5) + (threadIdx.x >> 5);
    const int lane = threadIdx.x & 31;
    const int mt = wave >> 4;        // 3125 row tiles
    const int nt = wave & 15;        // 16 col tiles
    if (mt * 16 >= NNODES) return;
    const int lanelo = lane & 15;
    const int h = lane >> 4;         // K-half select (K+0,1 vs K+2,3)
    // A[m][k] = emb[m*CH + k]; B[k][n] = w[n*CH + k]  -> both contiguous v2f
    const float* arow = emb + (long)(mt * 16 + lanelo) * CH + 2 * h;
    const float* brow = w   + (long)(nt * 16 + lanelo) * CH + 2 * h;
    v8f acc = {};
#pragma unroll 4
    for (int k = 0; k < CH; k += 4) {
        v2f a = *(const v2f*)(arow + k);
        v2f b = *(const v2f*)(brow + k);
        acc = __builtin_amdgcn_wmma_f32_16x16x4_f32(
            false, a, false, b, (short)0, acc, false, false);
    }
    // D layout: VGPR r holds M = r + 8*h, N = nt*16 + lanelo
    float* out = x + (long)(mt * 16 + 8 * h) * CH + nt * 16 + lanelo;
#pragma unroll
    for (int r = 0; r < 8; ++r) out[(long)r * CH] = acc[r];
}

// ---- e += x[node_idx]  (64 threads per incidence, float4 per thread) -------
__global__ __launch_bounds__(256) void edge_accum_k(const float* __restrict__ x,
                                                    const int* __restrict__ ni,
                                                    const int* __restrict__ ei,
                                                    float* __restrict__ e) {
    long t = (long)blockIdx.x * blockDim.x + threadIdx.x;
    int inc = (int)(t >> 6);
    if (inc >= NNZ) return;
    int c = ((int)t & 63) << 2;
    const float4 v = *(const float4*)(x + (long)ni[inc] * CH + c);
    float* dst = e + (long)ei[inc] * CH + c;
    atomicAdd(dst + 0, v.x);
    atomicAdd(dst + 1, v.y);
    atomicAdd(dst + 2, v.z);
    atomicAdd(dst + 3, v.w);
}

// ---- e *= Binv -------------------------------------------------------------
__global__ __launch_bounds__(256) void edge_scale_k(float* __restrict__ e,
                                                    const float* __restrict__ B) {
    int i = blockIdx.x * blockDim.x + threadIdx.x;
    int row = i >> 8;
    if (row < NEDGES) {
        float b = B[row];
        e[i] *= (b > 0.0f ? 1.0f / b : 0.0f);
    }
}

// ---- n += e[edge_idx] ------------------------------------------------------
__global__ __launch_bounds__(256) void node_accum_k(const float* __restrict__ e,
                                                    const int* __restrict__ ni,
                                                    const int* __restrict__ ei,
                                                    float* __restrict__ n) {
    long t = (long)blockIdx.x * blockDim.x + threadIdx.x;
    int inc = (int)(t >> 6);
    if (inc >= NNZ) return;
    int c = ((int)t & 63) << 2;
    const float4 v = *(const float4*)(e + (long)ei[inc] * CH + c);
    float* dst = n + (long)ni[inc] * CH + c;
    atomicAdd(dst + 0, v.x);
    atomicAdd(dst + 1, v.y);
    atomicAdd(dst + 2, v.z);
    atomicAdd(dst + 3, v.w);
}

// ---- y = leaky(n * Dinv + conv_b)  (in place over n) -----------------------
__global__ __launch_bounds__(256) void node_finish_k(float* __restrict__ n,
                                                     const float* __restrict__ D,
                                                     const float* __restrict__ cb) {
    int i = blockIdx.x * blockDim.x + threadIdx.x;
    int row = i >> 8;
    int c = i & 255;
    if (row < NNODES) {
        float d = D[row];
        float v = n[i] * (d > 0.0f ? 1.0f / d : 0.0f) + cb[c];
        n[i] = leaky(v);
    }
}

// ---- g = y^T y  LDS-staged strip GEMM --------------------------------------
// Block = 8 waves, owns output strip mt (16 rows) x all 256 cols for one
// K-chunk of 2000 rows.  Stage y[k..k+40][0:256] in LDS (coalesced float4
// global loads), each wave computes 2 column tiles reusing its A fragment.
// Register accumulation across the whole chunk; one atomic reduce at end.
__global__ __launch_bounds__(256) void gram_k(const float* __restrict__ y,
                                              float* __restrict__ g) {
    __shared__ float ly[GRAM_KT * CH];          // 40 KB

    const int mt   = blockIdx.x;                // 0..15
    const int k0   = blockIdx.y * GRAM_KCHUNK;  // chunk base row
    const int tid  = threadIdx.x;
    const int wave = tid >> 5;                  // 0..7
    const int lane = tid & 31;
    const int lanelo = lane & 15;
    const int h      = lane >> 4;               // K-half select
    const int mbase  = mt * 16;
    const int n0base = (wave * 2 + 0) * 16;
    const int n1base = (wave * 2 + 1) * 16;

    v8f acc0 = {};
    v8f acc1 = {};

    for (int kt = 0; kt < GRAM_KCHUNK; kt += GRAM_KT) {
        // stage 40 rows x 256 cols of y into LDS (b128, fully coalesced)
        const float4* src = (const float4*)(y + (long)(k0 + kt) * CH);
        float4* dst = (float4*)ly;
#pragma unroll
        for (int s = 0; s < GRAM_KT * (CH / 4) / 256; ++s)
            dst[tid + s * 256] = src[tid + s * 256];
        __syncthreads();

#pragma unroll 2
        for (int kk = 0; kk < GRAM_KT; kk += 4) {
            const float* r0 = ly + (kk + 2 * h) * CH;
            const float* r1 = r0 + CH;
            v2f a, b0, b1;
            a.x  = r0[mbase + lanelo];   a.y  = r1[mbase + lanelo];
            b0.x = r0[n0base + lanelo];  b0.y = r1[n0base + lanelo];
            b1.x = r0[n1base + lanelo];  b1.y = r1[n1base + lanelo];
            acc0 = __builtin_amdgcn_wmma_f32_16x16x4_f32(
                false, a, false, b0, (short)0, acc0, false, false);
            acc1 = __builtin_amdgcn_wmma_f32_16x16x4_f32(
                false, a, false, b1, (short)0, acc1, false, false);
        }
        __syncthreads();
    }

    // atomic split-K reduction into g (25 chunks, L2-resident)
    float* g0 = g + (long)(mbase + 8 * h) * CH + n0base + lanelo;
    float* g1 = g + (long)(mbase + 8 * h) * CH + n1base + lanelo;
#pragma unroll
    for (int r = 0; r < 8; ++r) {
        atomicAdd(&g0[(long)r * CH], acc0[r]);
        atomicAdd(&g1[(long)r * CH], acc1[r]);
    }
}

// ---- out = leaky(g @ lin_w^T + lin_b) --------------------------------------
__global__ __launch_bounds__(256) void final_k(const float* __restrict__ g,
                                               const float* __restrict__ lw,
                                               const float* __restrict__ lb,
                                               float* __restrict__ out) {
    const int wave = blockIdx.x * (blockDim.x >> 5) + (threadIdx.x >> 5);
    const int lane = threadIdx.x & 31;
    const int mt = wave >> 4;
    const int nt = wave & 15;
    const int lanelo = lane & 15;
    const int h = lane >> 4;
    const float* arow = g  + (long)(mt * 16 + lanelo) * CH + 2 * h;
    const float* brow = lw + (long)(nt * 16 + lanelo) * CH + 2 * h;
    v8f acc = {};
#pragma unroll 4
    for (int k = 0; k < CH; k += 4) {
        v2f a = *(const v2f*)(arow + k);
        v2f b = *(const v2f*)(brow + k);
        acc = __builtin_amdgcn_wmma_f32_16x16x4_f32(
            false, a, false, b, (short)0, acc, false, false);
    }
    const float bias = lb[nt * 16 + lanelo];
    float* op = out + (long)(mt * 16 + 8 * h) * CH + nt * 16 + lanelo;
#pragma unroll
    for (int r = 0; r < 8; ++r) op[(long)r * CH] = leaky(acc[r] + bias);
}

// ---------------------------------------------------------------------------
extern "C" void kernel_launch(void* const* d_in, const int* in_sizes, int n_in,
                              void* d_out, int out_size, void* d_ws, size_t ws_size,
                              hipStream_t stream) {
    const float* emb    = (const float*)d_in[0];   // [50000,256]
    const float* conv_w = (const float*)d_in[1];   // [256,256]
    const float* conv_b = (const float*)d_in[2];   // [256]
    const float* lin_w  = (const float*)d_in[3];   // [256,256]
    const float* lin_b  = (const float*)d_in[4];   // [256]
    const int*   eidx   = (const int*)d_in[5];     // [2,800000]
    const int* node_idx = eidx;
    const int* edge_idx = eidx + NNZ;
    float* out = (float*)d_out;

    // Workspace layout (floats)
    float* ws   = (float*)d_ws;
    float* xbuf = ws;                                     // 50000*256
    float* nbuf = xbuf + (long)NNODES * CH;               // 50000*256
    float* ebuf = nbuf + (long)NNODES * CH;               // 10000*256
    float* Ddeg = ebuf + (long)NEDGES * CH;               // 50000
    float* Bdeg = Ddeg + NNODES;                          // 10000
    float* gbuf = Bdeg + NEDGES;                          // 256*256
    const long zero_count = (long)NNODES * CH + (long)NEDGES * CH
                          + NNODES + NEDGES + (long)CH * CH;

    // 1. zero everything that is atomically accumulated (n, e, D, B, g)
    zero_f<<<4096, 256, 0, stream>>>(nbuf, zero_count);

    // 2. degree histograms
    degrees_k<<<(NNZ + 255) / 256, 256, 0, stream>>>(node_idx, edge_idx, Ddeg, Bdeg);

    // 3. x = emb @ conv_w^T   (50000 tile-waves, 8 waves/block)
    gemm_xw_k<<<(NNODES / 16) * 16 / 8, 256, 0, stream>>>(emb, conv_w, xbuf);

    // 4. e scatter + scale
    edge_accum_k<<<(long)NNZ * 64 / 256, 256, 0, stream>>>(xbuf, node_idx, edge_idx, ebuf);
    edge_scale_k<<<(NEDGES * CH) / 256, 256, 0, stream>>>(ebuf, Bdeg);

    // 5. n scatter + finish (y = leaky(n/D + cb), in place)
    node_accum_k<<<(long)NNZ * 64 / 256, 256, 0, stream>>>(ebuf, node_idx, edge_idx, nbuf);
    node_finish_k<<<(NNODES * CH) / 256, 256, 0, stream>>>(nbuf, Ddeg, conv_b);

    // 6. g = y^T y   (16 strips x 25 K-chunks, 8 waves/block, LDS-staged)
    gram_k<<<dim3(16, GRAM_CHUNKS), 256, 0, stream>>>(nbuf, gbuf);

    // 7. out = leaky(g @ lin_w^T + lin_b)
    final_k<<<32, 256, 0, stream>>>(gbuf, lin_w, lin_b, out);
}